// MoERouter_22411139350727
// MI455X (gfx1250) — compile-verified
//
#include <hip/hip_runtime.h>
#include <hip/hip_bf16.h>
#include <math.h>

// MoE router: logits = x @ W^T + b ; softmax -> top2 -> normalized weights,
// indices, one-hot mask permuted to (E, K, N).
//
// Shapes: x (16384, 2048) f32, W (64, 2048) f32, b (64,) f32.
// d_out layout (flat f32, reference tuple order):
//   [0)                 logits   N*E
//   [N*E)               weights  N*2
//   [N*E + 2N)          indices  N*2   (integer values stored as floats)
//   [N*E + 4N)          mask     E*2*N (0.0 / 1.0)

#define HIDDEN  2048
#define EXPERTS 64
#define NROWS   16384

#define KC  128          // K-slice staged in LDS per pipeline stage
#define KCP 132          // padded LDS row stride (dwords): 16B-aligned, bank-spread
#define NCHUNK (HIDDEN / KC)

typedef float v2f __attribute__((ext_vector_type(2)));
typedef float v8f __attribute__((ext_vector_type(8)));

// Raw 32-bit LDS offset from a generic pointer to a __shared__ object:
// per CDNA5 aperture rules, the LDS offset is the low 32 bits of the address.
__device__ __forceinline__ unsigned lds_off(const void* p) {
  return (unsigned)(unsigned long long)p;
}

// Async-stage one 64xKC slice of W into an LDS buffer (tracked by ASYNCcnt).
// Closed-form indexing: issue i covers expert row e = 8*i + wave, j = 4*lane,
// i.e. per-issue strides of 8*HIDDEN*4 bytes (global) / 8*KCP*4 bytes (LDS).
// saddr (GVS) form: uniform W base rides in SGPRs, per-lane 32-bit voffset.
__device__ __forceinline__ void stage_w_async(const float* __restrict__ W,
                                              unsigned gvoff, unsigned lbase) {
#pragma unroll
  for (int i = 0; i < (EXPERTS * KC) / (256 * 4); ++i) {
    unsigned v = gvoff + (unsigned)(i * 8 * HIDDEN * 4);
    unsigned l = lbase + (unsigned)(i * 8 * KCP * 4);   // 16B aligned
    asm volatile("global_load_async_to_lds_b128 %0, %1, %2"
                 :: "v"(l), "v"(v), "s"(W) : "memory");
  }
}

// ---------------------------------------------------------------------------
// GEMM: one wave computes a 16-row x 64-expert tile via V_WMMA_F32_16X16X4_F32.
// Block = 8 waves = 128 rows. Grid = NROWS/128 = 128 blocks.
// W slices are double-buffered in LDS via async global->LDS copies.
// ---------------------------------------------------------------------------
__global__ __launch_bounds__(256) void moe_router_gemm(
    const float* __restrict__ x, const float* __restrict__ W,
    const float* __restrict__ b, float* __restrict__ logits)
{
  __shared__ float lw[2][EXPERTS * KCP];   // 2 x 33792 B

  const int tid  = threadIdx.x;
  const int lane = tid & 31;
  const int wave = tid >> 5;
  const int n    = lane & 15;   // column within 16-wide tile / row M for A
  const int hi   = lane >> 4;   // 0 -> K pair {0,1}, 1 -> K pair {2,3}
  const int rowBase = blockIdx.x * 128 + wave * 16;

  // Per-thread staging bases (bytes).
  const unsigned gvbase = (unsigned)((wave * HIDDEN + lane * 4) * 4);
  const unsigned lbase0 = lds_off(&lw[0][0]) + (unsigned)((wave * KCP + lane * 4) * 4);
  const unsigned lbase1 = lds_off(&lw[1][0]) + (unsigned)((wave * KCP + lane * 4) * 4);

  v8f acc0 = {}, acc1 = {}, acc2 = {}, acc3 = {};

  // A-matrix 16x4 f32 layout: lanes 0-15 hold row n K{0,1}; lanes 16-31 K{2,3}
  const float* xrow = x + (size_t)(rowBase + n) * HIDDEN + hi * 2;

  // Prologue: stage chunk 0.
  stage_w_async(W, gvbase, lbase0);

  for (int c = 0; c < NCHUNK; ++c) {
    // This wave's async stores for buf c&1 done, then collective visibility.
    asm volatile("s_wait_asynccnt 0x0" ::: "memory");
    __syncthreads();

    // Stage next chunk into the other buffer; its previous readers (chunk c-1
    // compute) all finished before the barrier above.
    if (c + 1 < NCHUNK)
      stage_w_async(W, gvbase + (unsigned)((c + 1) * KC * 4),
                    ((c + 1) & 1) ? lbase1 : lbase0);

    const float* Bs = &lw[c & 1][0];
    const int k0 = c * KC;

#pragma unroll 8
    for (int s = 0; s < KC / 4; ++s) {
      v2f a = *(const v2f*)(xrow + k0 + 4 * s);
      // B-matrix 4x16 f32: lane(n,hi) holds col e = t*16+n, rows K{2hi, 2hi+1}
      const int kk = 4 * s + hi * 2;
      v2f b0 = *(const v2f*)(Bs + (0 * 16 + n) * KCP + kk);
      v2f b1 = *(const v2f*)(Bs + (1 * 16 + n) * KCP + kk);
      v2f b2 = *(const v2f*)(Bs + (2 * 16 + n) * KCP + kk);
      v2f b3 = *(const v2f*)(Bs + (3 * 16 + n) * KCP + kk);
      acc0 = __builtin_amdgcn_wmma_f32_16x16x4_f32(false, a, false, b0,
                                                   (short)0, acc0, false, false);
      acc1 = __builtin_amdgcn_wmma_f32_16x16x4_f32(false, a, false, b1,
                                                   (short)0, acc1, false, false);
      acc2 = __builtin_amdgcn_wmma_f32_16x16x4_f32(false, a, false, b2,
                                                   (short)0, acc2, false, false);
      acc3 = __builtin_amdgcn_wmma_f32_16x16x4_f32(false, a, false, b3,
                                                   (short)0, acc3, false, false);
    }
  }

  // Bias per expert column.
  float bias0 = b[0 * 16 + n];
  float bias1 = b[1 * 16 + n];
  float bias2 = b[2 * 16 + n];
  float bias3 = b[3 * 16 + n];

  // C/D layout: VGPR v, lanes 0-15 -> row v, lanes 16-31 -> row v+8; col = lane%16.
  float* outp = logits + (size_t)(rowBase + hi * 8) * EXPERTS + n;
#pragma unroll
  for (int v = 0; v < 8; ++v) {
    size_t ro = (size_t)v * EXPERTS;
    outp[ro + 0 * 16] = acc0[v] + bias0;
    outp[ro + 1 * 16] = acc1[v] + bias1;
    outp[ro + 2 * 16] = acc2[v] + bias2;
    outp[ro + 3 * 16] = acc3[v] + bias3;
  }
}

// ---------------------------------------------------------------------------
// Zero-fill the (E, K, N) mask region, coalesced float4 stores.
// ---------------------------------------------------------------------------
__global__ __launch_bounds__(256) void moe_mask_zero(float4* __restrict__ p, int count4)
{
  int i = blockIdx.x * blockDim.x + threadIdx.x;
  if (i < count4) p[i] = make_float4(0.f, 0.f, 0.f, 0.f);
}

// ---------------------------------------------------------------------------
// Per-row top-2 + normalized weights + index/mask scatter.
// Softmax denominator cancels: w1 = 1/(1+e^{l2-l1}), w2 = e^{l2-l1}/(1+e^{l2-l1}).
// ---------------------------------------------------------------------------
__global__ __launch_bounds__(256) void moe_router_topk(
    const float* __restrict__ logits, float* __restrict__ weights,
    float* __restrict__ indices, float* __restrict__ mask)
{
  int row = blockIdx.x * blockDim.x + threadIdx.x;
  if (row >= NROWS) return;

  const float4* l4 = (const float4*)(logits + (size_t)row * EXPERTS);
  float m1 = -INFINITY, m2 = -INFINITY;
  int i1 = 0, i2 = 0;
#pragma unroll
  for (int q = 0; q < EXPERTS / 4; ++q) {
    float4 v4 = l4[q];
    float vv[4] = {v4.x, v4.y, v4.z, v4.w};
#pragma unroll
    for (int t = 0; t < 4; ++t) {
      float v = vv[t];
      int e = q * 4 + t;
      // strict > keeps first occurrence on ties, matching jax.lax.top_k
      if (v > m1) { m2 = m1; i2 = i1; m1 = v; i1 = e; }
      else if (v > m2) { m2 = v; i2 = e; }
    }
  }
  float e2  = __expf(m2 - m1);
  float rcp = 1.0f / (1.0f + e2);
  weights[row * 2 + 0] = rcp;
  weights[row * 2 + 1] = e2 * rcp;
  indices[row * 2 + 0] = (float)i1;
  indices[row * 2 + 1] = (float)i2;
  // mask (E, K, N): mask[e][k][n]
  mask[(size_t)i1 * (2 * NROWS) + 0 * NROWS + row] = 1.0f;
  mask[(size_t)i2 * (2 * NROWS) + 1 * NROWS + row] = 1.0f;
}

// ---------------------------------------------------------------------------
extern "C" void kernel_launch(void* const* d_in, const int* in_sizes, int n_in,
                              void* d_out, int out_size, void* d_ws, size_t ws_size,
                              hipStream_t stream) {
  const float* x = (const float*)d_in[0];
  const float* W = (const float*)d_in[1];
  const float* b = (const float*)d_in[2];

  float* out     = (float*)d_out;
  float* logits  = out;                                   // N*E
  float* weights = logits + (size_t)NROWS * EXPERTS;      // N*2
  float* indices = weights + (size_t)NROWS * 2;           // N*2
  float* mask    = indices + (size_t)NROWS * 2;           // E*2*N

  moe_router_gemm<<<NROWS / 128, 256, 0, stream>>>(x, W, b, logits);

  int maskCount4 = (EXPERTS * 2 * NROWS) / 4;
  moe_mask_zero<<<(maskCount4 + 255) / 256, 256, 0, stream>>>((float4*)mask,
                                                              maskCount4);

  moe_router_topk<<<NROWS / 256, 256, 0, stream>>>(logits, weights, indices,
                                                   mask);
}